// ProbMambaHead_43851616092544
// MI455X (gfx1250) — compile-verified
//
#include <hip/hip_runtime.h>
#include <hip/hip_bf16.h>
#include <math.h>

typedef __attribute__((ext_vector_type(2))) float v2f;
typedef __attribute__((ext_vector_type(8))) float v8f;

#define Bsz 32
#define Tsz 2048
#define DF  128
#define NST 16
#define DYV 8
#define BT  (Bsz * Tsz)
#define RWG 64          // rows of (B*T) per workgroup (4 x 16-row WMMA tiles)
#define XPITCH 132      // LDS row pitch (16B aligned, conflict-free)

__device__ __forceinline__ float sp_(float x) {            // softplus
  return (x > 20.f) ? x : log1pf(expf(x));
}
__device__ __forceinline__ float phi_(float z, float k) {  // (exp(kz)-1)/(kz)
  float kz = k * z;
  if (fabsf(kz) < 1e-4f) return 1.f + 0.5f * kz;
  return expm1f(kz) / kz;
}

// ---------------------------------------------------------------------------
// Kernel 1: per-(b,t) SSM params. One wave handles 64 rows of (B*T); each
// global W-fragment (b64) feeds 4 V_WMMA_F32_16X16X4_F32 (one per row-tile),
// cutting L2 traffic for W_B by 4x. u = x@W_B^T contracted on the fly into hx.
// ---------------------------------------------------------------------------
__global__ __launch_bounds__(32) void param_kernel(
    const float* __restrict__ x_feat, const float* __restrict__ a_raw,
    const float* __restrict__ W_gate, const float* __restrict__ b_gate,
    const float* __restrict__ W_B,   const float* __restrict__ b_B,
    const float* __restrict__ W_C,   const float* __restrict__ b_C,
    const float* __restrict__ W_sig, const float* __restrict__ b_sig,
    const float* __restrict__ W_R,   const float* __restrict__ b_R,
    float* __restrict__ expz_ws, float* __restrict__ hx_ws,
    float* __restrict__ q_ws, float* __restrict__ ck_ws,
    float* __restrict__ rk_ws)
{
  __shared__ float xl[RWG][XPITCH];
  __shared__ float hxacc[RWG][16];
  __shared__ float delta_l[RWG];
  __shared__ float sig_l[RWG][16];

  const int lane   = threadIdx.x;
  const long r0    = (long)blockIdx.x * RWG;
  const int  lm    = lane & 15;
  const int  halfu = lane >> 4;             // 0|1
  const int  koff  = halfu * 2;             // ISA frag layout: K-offset per half

  // stage x block (64 x 128) via float4
  {
    const float4* x4 = (const float4*)(x_feat + r0 * DF);
    for (int i = lane; i < RWG * (DF / 4); i += 32) {
      const int row = i >> 5, c = (i & 31) * 4;
      const float4 v = x4[i];
      xl[row][c] = v.x; xl[row][c + 1] = v.y; xl[row][c + 2] = v.z; xl[row][c + 3] = v.w;
    }
  }
  for (int i = lane; i < RWG * 16; i += 32) hxacc[i >> 4][i & 15] = 0.f;
  __syncthreads();

  // 128 column tiles of u (W_B) + 8 tiles of Ck (W_C)
  for (int tile = 0; tile < 136; ++tile) {
    const bool isB = (tile < 128);
    const int c0   = isB ? tile * 16 : (tile - 128) * 16;
    const float* W = isB ? W_B : W_C;
    const int col  = c0 + lm;
    v8f acc0 = {}, acc1 = {}, acc2 = {}, acc3 = {};
#pragma unroll 4
    for (int kk = 0; kk < DF; kk += 4) {
      v2f bf;
      bf.x = W[(long)col * DF + kk + koff];
      bf.y = W[(long)col * DF + kk + koff + 1];
      v2f a0, a1, a2, a3;
      a0.x = xl[lm][kk + koff];      a0.y = xl[lm][kk + koff + 1];
      a1.x = xl[16 + lm][kk + koff]; a1.y = xl[16 + lm][kk + koff + 1];
      a2.x = xl[32 + lm][kk + koff]; a2.y = xl[32 + lm][kk + koff + 1];
      a3.x = xl[48 + lm][kk + koff]; a3.y = xl[48 + lm][kk + koff + 1];
      acc0 = __builtin_amdgcn_wmma_f32_16x16x4_f32(false, a0, false, bf, (short)0, acc0, false, false);
      acc1 = __builtin_amdgcn_wmma_f32_16x16x4_f32(false, a1, false, bf, (short)0, acc1, false, false);
      acc2 = __builtin_amdgcn_wmma_f32_16x16x4_f32(false, a2, false, bf, (short)0, acc2, false, false);
      acc3 = __builtin_amdgcn_wmma_f32_16x16x4_f32(false, a3, false, bf, (short)0, acc3, false, false);
    }
    if (isB) {
      const int j  = tile >> 3;
      const int fb = (tile & 7) * 16;
      const float bb = b_B[col];
#pragma unroll
      for (int rb = 0; rb < 4; ++rb) {
        const v8f acc = (rb == 0) ? acc0 : (rb == 1) ? acc1 : (rb == 2) ? acc2 : acc3;
#pragma unroll
        for (int v = 0; v < 8; ++v) {
          const int row = rb * 16 + v + halfu * 8;   // C/D layout: vgpr v -> row v/v+8
          float val = (acc[v] + bb) * xl[row][fb + lm];
          val += __shfl_xor(val, 1);
          val += __shfl_xor(val, 2);
          val += __shfl_xor(val, 4);
          val += __shfl_xor(val, 8);                 // sum 16 cols within half-wave
          if (lm == 0) hxacc[row][j] += val;         // lanes 0 & 16: distinct rows
        }
      }
    } else {
      const float bc = b_C[col];
#pragma unroll
      for (int rb = 0; rb < 4; ++rb) {
        const v8f acc = (rb == 0) ? acc0 : (rb == 1) ? acc1 : (rb == 2) ? acc2 : acc3;
#pragma unroll
        for (int v = 0; v < 8; ++v) {
          const int row = rb * 16 + v + halfu * 8;
          ck_ws[(r0 + row) * 128 + col] = acc[v] + bc;
        }
      }
    }
  }
  __syncthreads();

  // small projections: 64 rows x 25 outputs (gate=0, sig=1..16, R=17..24)
  for (int item = lane; item < RWG * 25; item += 32) {
    const int row = item / 25, o = item % 25;
    const float* w;
    if (o == 0)       w = W_gate;
    else if (o <= 16) w = W_sig + (o - 1) * DF;
    else              w = W_R + (o - 17) * DF;
    const float4* w4 = (const float4*)w;
    const float4* x4 = (const float4*)&xl[row][0];
    float s = 0.f;
#pragma unroll 8
    for (int f = 0; f < DF / 4; ++f) {
      const float4 a = x4[f], b = w4[f];
      s += a.x * b.x + a.y * b.y + a.z * b.z + a.w * b.w;
    }
    if (o == 0) {
      float d = sp_(s + b_gate[0]) + 1e-6f;
      delta_l[row] = fminf(fmaxf(d, 1e-3f), 1.0f);
    } else if (o <= 16) {
      sig_l[row][o - 1] = sp_(s + b_sig[o - 1]) + 1e-6f + 1e-3f;
    } else {
      const int d = o - 17;
      rk_ws[(r0 + row) * 8 + d] = sp_(s + b_R[d]) + 1e-6f + 1e-4f;
    }
  }
  __syncthreads();

  for (int idx = lane; idx < RWG * 16; idx += 32) {
    const int row = idx >> 4, j = idx & 15;
    const float d = delta_l[row];
    const float a = -(sp_(a_raw[j]) + 1e-6f);
    float z = fminf(fmaxf(d * a, -20.f), 20.f);
    const long o = (r0 + row) * 16 + j;
    expz_ws[o] = expf(z);
    hx_ws[o]   = phi_(z, 1.f) * d * hxacc[row][j];
    const float sg = sig_l[row][j];
    q_ws[o] = sg * sg * phi_(z, 2.f) * d;
  }
}

// ---------------------------------------------------------------------------
// Kernel 2: Kalman scan. One wave32 per batch; 16 f32 WMMAs per step for the
// 16x16x16 products; 8x8 Cholesky serial on lane 0; 17 triangular solves
// (16 CP columns + innovation) one per lane. Next-step params prefetched.
// ---------------------------------------------------------------------------
__global__ __launch_bounds__(32) void scan_kernel(
    const float* __restrict__ y, const float* __restrict__ p0,
    const float* __restrict__ expz_ws, const float* __restrict__ hx_ws,
    const float* __restrict__ q_ws, const float* __restrict__ ck_ws,
    const float* __restrict__ rk_ws,
    float* __restrict__ y_mean, float* __restrict__ y_var,
    float* __restrict__ llsum)
{
  __shared__ float P[16][17], Pp[16][17], Cp[16][17], CP[16][17];
  __shared__ float T1[16][17], IK[16][17], S[16][17];
  __shared__ float Km[16][8], Lc[8][9];
  __shared__ float ez[16], hx[16], qv[16], hst[16], hp[16];
  __shared__ float Rt[8], yt8[8], ev[8];
  __shared__ float logdet_s, quad_s;

  const int lane  = threadIdx.x;
  const int b     = blockIdx.x;
  const int lm    = lane & 15;
  const int halfu = lane >> 4;
  const int koff  = halfu * 2;

  for (int i = lane; i < 256; i += 32) { P[i >> 4][i & 15] = 0.f; Cp[i >> 4][i & 15] = 0.f; }
  if (lane < 16) { P[lane][lane] = fabsf(p0[lane]); hst[lane] = 0.f; }
  float llacc = 0.f;
  __syncthreads();

  for (int t = 0; t < Tsz; ++t) {
    const long base = (long)b * Tsz + t;
    // prefetch next step's parameter lines (global_prefetch_b8)
    if (t + 1 < Tsz) {
      __builtin_prefetch(ck_ws + (base + 1) * 128 + lane * 4, 0, 1);
      __builtin_prefetch(expz_ws + (base + 1) * 16 + lane, 0, 1);
    }
    if (lane < 16) {
      ez[lane] = expz_ws[base * 16 + lane];
      hx[lane] = hx_ws[base * 16 + lane];
      qv[lane] = q_ws[base * 16 + lane];
    } else if (lane < 24) {
      const int d = lane - 16;
      Rt[d] = rk_ws[base * 8 + d];
      yt8[d] = y[base * 8 + d];
    }
    {  // Ck tile (8x16), lane covers 4 consecutive cols of row lane>>2
      const float4 c4 = ((const float4*)(ck_ws + base * 128))[lane];
      const int d = lane >> 2, j0 = (lane & 3) * 4;
      Cp[d][j0 + 0] = c4.x; Cp[d][j0 + 1] = c4.y;
      Cp[d][j0 + 2] = c4.z; Cp[d][j0 + 3] = c4.w;
    }
    __syncthreads();

    if (lane < 16) hp[lane] = ez[lane] * hst[lane] + hx[lane];
#pragma unroll
    for (int v = 0; v < 8; ++v) {   // P_pred = diag(ez) P diag(ez) + diag(q)
      const int i = v + halfu * 8;
      float pv = ez[i] * ez[lm] * P[i][lm];
      if (i == lm) pv += qv[i];
      Pp[i][lm] = pv;
    }
    __syncthreads();

    {  // CP = Cpad @ Ppred   (4 WMMAs)
      v8f acc = {};
#pragma unroll
      for (int kk = 0; kk < 16; kk += 4) {
        v2f a, bf;
        a.x = Cp[lm][kk + koff];      a.y = Cp[lm][kk + koff + 1];
        bf.x = Pp[kk + koff][lm];     bf.y = Pp[kk + koff + 1][lm];
        acc = __builtin_amdgcn_wmma_f32_16x16x4_f32(false, a, false, bf,
                                                    (short)0, acc, false, false);
      }
#pragma unroll
      for (int v = 0; v < 8; ++v) CP[v + halfu * 8][lm] = acc[v];
    }
    __syncthreads();
    if (lane < 8) {                 // y_pred, innovation
      float yp = 0.f;
      for (int j = 0; j < 16; ++j) yp += Cp[lane][j] * hp[j];
      y_mean[base * 8 + lane] = yp;
      ev[lane] = yt8[lane] - yp;
    }
    __syncthreads();

    {  // S = CP @ Cpad^T   (4 WMMAs, B read transposed)
      v8f acc = {};
#pragma unroll
      for (int kk = 0; kk < 16; kk += 4) {
        v2f a, bf;
        a.x = CP[lm][kk + koff];  a.y = CP[lm][kk + koff + 1];
        bf.x = Cp[lm][kk + koff]; bf.y = Cp[lm][kk + koff + 1];
        acc = __builtin_amdgcn_wmma_f32_16x16x4_f32(false, a, false, bf,
                                                    (short)0, acc, false, false);
      }
#pragma unroll
      for (int v = 0; v < 8; ++v) S[v + halfu * 8][lm] = acc[v];
    }
    __syncthreads();
    if (lane < 8) {
      S[lane][lane] += Rt[lane];
      y_var[base * 8 + lane] = S[lane][lane];
    }
    __syncthreads();

    if (lane == 0) {                // 8x8 Cholesky, serial
      float ld = 0.f;
      for (int k2 = 0; k2 < 8; ++k2) {
        float dcc = S[k2][k2];
        for (int m = 0; m < k2; ++m) dcc -= Lc[k2][m] * Lc[k2][m];
        dcc = sqrtf(fmaxf(dcc, 1e-20f));
        Lc[k2][k2] = dcc; ld += logf(dcc);
        const float inv = 1.f / dcc;
        for (int i2 = k2 + 1; i2 < 8; ++i2) {
          float s2 = S[i2][k2];
          for (int m = 0; m < k2; ++m) s2 -= Lc[i2][m] * Lc[k2][m];
          Lc[i2][k2] = s2 * inv;
        }
      }
      logdet_s = 2.f * ld;
    }
    __syncthreads();

    if (lane < 17) {                // solves: 16 CP columns + e
      float rhs[8], yv[8], xv[8];
      for (int d = 0; d < 8; ++d) rhs[d] = (lane < 16) ? CP[d][lane] : ev[d];
      for (int d = 0; d < 8; ++d) {
        float s2 = rhs[d];
        for (int m = 0; m < d; ++m) s2 -= Lc[d][m] * yv[m];
        yv[d] = s2 / Lc[d][d];
      }
      for (int d = 7; d >= 0; --d) {
        float s2 = yv[d];
        for (int m = d + 1; m < 8; ++m) s2 -= Lc[m][d] * xv[m];
        xv[d] = s2 / Lc[d][d];
      }
      if (lane < 16) {
        for (int d = 0; d < 8; ++d) Km[lane][d] = xv[d];   // K = X^T
      } else {
        float qd = 0.f;
        for (int d = 0; d < 8; ++d) qd += ev[d] * xv[d];
        quad_s = qd;
      }
    }
    __syncthreads();
    if (lane == 16)
      llacc += -0.5f * (logdet_s + quad_s + 8.f * 1.8378770664093453f);

    if (lane < 16) {                // h_new = h_pred + K e
      float s2 = hp[lane];
      for (int d = 0; d < 8; ++d) s2 += Km[lane][d] * ev[d];
      hst[lane] = s2;
    }
#pragma unroll
    for (int v = 0; v < 8; ++v) {   // I - K C
      const int i = v + halfu * 8;
      float s2 = (i == lm) ? 1.f : 0.f;
      for (int d = 0; d < 8; ++d) s2 -= Km[i][d] * Cp[d][lm];
      IK[i][lm] = s2;
    }
    __syncthreads();

    {  // T1 = IK @ Ppred
      v8f acc = {};
#pragma unroll
      for (int kk = 0; kk < 16; kk += 4) {
        v2f a, bf;
        a.x = IK[lm][kk + koff];  a.y = IK[lm][kk + koff + 1];
        bf.x = Pp[kk + koff][lm]; bf.y = Pp[kk + koff + 1][lm];
        acc = __builtin_amdgcn_wmma_f32_16x16x4_f32(false, a, false, bf,
                                                    (short)0, acc, false, false);
      }
#pragma unroll
      for (int v = 0; v < 8; ++v) T1[v + halfu * 8][lm] = acc[v];
    }
    __syncthreads();

    {  // P_new = T1 @ IK^T + K R K^T
      v8f acc = {};
#pragma unroll
      for (int kk = 0; kk < 16; kk += 4) {
        v2f a, bf;
        a.x = T1[lm][kk + koff];  a.y = T1[lm][kk + koff + 1];
        bf.x = IK[lm][kk + koff]; bf.y = IK[lm][kk + koff + 1];  // IK^T
        acc = __builtin_amdgcn_wmma_f32_16x16x4_f32(false, a, false, bf,
                                                    (short)0, acc, false, false);
      }
      __syncthreads();
#pragma unroll
      for (int v = 0; v < 8; ++v) {
        const int i = v + halfu * 8;
        float s2 = acc[v];
        for (int d = 0; d < 8; ++d) s2 += Km[i][d] * Rt[d] * Km[lm][d];
        P[i][lm] = s2;
      }
    }
    __syncthreads();
  }
  if (lane == 16) llsum[b] = llacc;
}

// ---------------------------------------------------------------------------
// Kernel 3: reduce 32 per-batch ll sums -> nll, avg_ll_per_step
// ---------------------------------------------------------------------------
__global__ __launch_bounds__(32) void finalize_kernel(
    const float* __restrict__ llsum, float* __restrict__ outp)
{
  float v = llsum[threadIdx.x];
  for (int off = 16; off > 0; off >>= 1) v += __shfl_xor(v, off);
  if (threadIdx.x == 0) {
    outp[0] = -v / (float)Bsz;                 // nll
    outp[1] = v / (float)(Bsz * Tsz);          // avg ll per step
  }
}

extern "C" void kernel_launch(void* const* d_in, const int* in_sizes, int n_in,
                              void* d_out, int out_size, void* d_ws, size_t ws_size,
                              hipStream_t stream) {
  const float* x_feat = (const float*)d_in[0];
  const float* y      = (const float*)d_in[1];
  const float* a_raw  = (const float*)d_in[2];
  const float* W_gate = (const float*)d_in[3];
  const float* b_gate = (const float*)d_in[4];
  const float* W_B    = (const float*)d_in[5];
  const float* b_B    = (const float*)d_in[6];
  const float* W_C    = (const float*)d_in[7];
  const float* b_C    = (const float*)d_in[8];
  const float* W_sig  = (const float*)d_in[9];
  const float* b_sig  = (const float*)d_in[10];
  const float* W_R    = (const float*)d_in[11];
  const float* b_R    = (const float*)d_in[12];
  const float* p0     = (const float*)d_in[13];

  float* ws = (float*)d_ws;
  float* expz_ws = ws;                              // BT*16
  float* hx_ws   = expz_ws + (size_t)BT * 16;       // BT*16
  float* q_ws    = hx_ws   + (size_t)BT * 16;       // BT*16
  float* ck_ws   = q_ws    + (size_t)BT * 16;       // BT*128
  float* rk_ws   = ck_ws   + (size_t)BT * 128;      // BT*8
  float* llsum   = rk_ws   + (size_t)BT * 8;        // 32

  float* out    = (float*)d_out;
  float* y_mean = out;                               // BT*8
  float* y_var  = out + (size_t)BT * 8;              // BT*8
  float* scal   = out + (size_t)2 * BT * 8;          // nll, avg_ll

  param_kernel<<<BT / RWG, 32, 0, stream>>>(
      x_feat, a_raw, W_gate, b_gate, W_B, b_B, W_C, b_C,
      W_sig, b_sig, W_R, b_R, expz_ws, hx_ws, q_ws, ck_ws, rk_ws);

  scan_kernel<<<Bsz, 32, 0, stream>>>(
      y, p0, expz_ws, hx_ws, q_ws, ck_ws, rk_ws, y_mean, y_var, llsum);

  finalize_kernel<<<1, 32, 0, stream>>>(llsum, scal);
}